// DistMul_81690277970461
// MI455X (gfx1250) — compile-verified
//
#include <hip/hip_runtime.h>
#include <math.h>

typedef __attribute__((ext_vector_type(2))) float v2f;
typedef __attribute__((ext_vector_type(8))) float v8f;

#define EMB   256
#define BATCH 2048
#define TILE  64
#define KC    32
#define KPAD  (KC + 4)   // 36-DWORD row stride: conflict-free (36*i mod 64 injective
                         // for i=0..15) and keeps float4 / b64 LDS accesses aligned

// ---------------------------------------------------------------------------
// Kernel 1: per-example hr[b][e] = sum_d heads[b][d] * rel[r_b][d][e]
//           plus gather of tails into a dense workspace buffer.
// One block per batch element, 256 threads (one per output column e).
// rel reads are coalesced (consecutive e across lanes) and L2-resident
// (500 * 256KB = 128MB < 192MB L2).
// ---------------------------------------------------------------------------
__global__ void __launch_bounds__(EMB)
hr_gather_kernel(const int* __restrict__ bh, const int* __restrict__ bt,
                 const int* __restrict__ br,
                 const float* __restrict__ ent, const float* __restrict__ rel,
                 float* __restrict__ hr, float* __restrict__ tails) {
    __shared__ float sh[EMB];
    const int b = blockIdx.x;
    const int e = threadIdx.x;
    const int h = bh[b];
    const int t = bt[b];
    const int r = br[b];

    sh[e] = ent[(long long)h * EMB + e];
    tails[b * EMB + e] = ent[(long long)t * EMB + e];
    __syncthreads();

    const float* __restrict__ R = rel + (long long)r * EMB * EMB;
    float acc = 0.0f;
#pragma unroll 8
    for (int d = 0; d < EMB; ++d) {
        acc = fmaf(sh[d], R[d * EMB + e], acc);
    }
    hr[b * EMB + e] = acc;
}

// ---------------------------------------------------------------------------
// Kernel 2: score = sigmoid(hr @ tails^T), 2048x2048x256 f32 GEMM via
// V_WMMA_F32_16X16X4_F32. 256 threads = 8 wave32 per block, 64x64 output
// tile, K staged through LDS in 32-wide chunks. Each wave owns two 16x16
// accumulator tiles (sharing the A fragment) -> 16 subtiles cover 64x64.
// ---------------------------------------------------------------------------
__global__ void __launch_bounds__(256)
score_wmma_kernel(const float* __restrict__ hr, const float* __restrict__ tails,
                  float* __restrict__ out) {
    __shared__ float As[TILE][KPAD];   // hr rows     (M x K chunk)
    __shared__ float Bs[TILE][KPAD];   // tails rows  (N x K chunk)

    const int tm   = blockIdx.x * TILE;
    const int tn   = blockIdx.y * TILE;
    const int tid  = threadIdx.x;
    const int lane = tid & 31;
    const int wave = tid >> 5;          // 0..7

    // subtile s = wave*2 + j; 4x4 grid of 16x16 subtiles.
    // s0 and s1 share subRow -> shared A fragment.
    const int s0row = ((wave * 2 + 0) >> 2) * 16;
    const int s0col = ((wave * 2 + 0) & 3) * 16;
    const int s1col = ((wave * 2 + 1) & 3) * 16;

    v8f c0 = {};
    v8f c1 = {};

    const int mr  = lane & 15;            // M (or N) index within subtile
    const int khi = (lane >> 4) << 1;     // lanes 16-31 handle K+2,K+3

    for (int kk = 0; kk < EMB; kk += KC) {
        // Cooperative staging: 64 rows x 32 cols = 512 float4, 2 per thread.
#pragma unroll
        for (int i = 0; i < 2; ++i) {
            const int f   = tid + i * 256;       // float4 index 0..511
            const int row = f >> 3;              // 8 float4 per 32-float row
            const int c4  = (f & 7) << 2;
            *(float4*)&As[row][c4] =
                *(const float4*)&hr[(tm + row) * EMB + kk + c4];
            *(float4*)&Bs[row][c4] =
                *(const float4*)&tails[(tn + row) * EMB + kk + c4];
        }
        __syncthreads();

#pragma unroll
        for (int k4 = 0; k4 < KC / 4; ++k4) {
            const int kb = k4 * 4 + khi;

            v2f a, b0, b1;
            a.x  = As[s0row + mr][kb];
            a.y  = As[s0row + mr][kb + 1];
            b0.x = Bs[s0col + mr][kb];
            b0.y = Bs[s0col + mr][kb + 1];
            b1.x = Bs[s1col + mr][kb];
            b1.y = Bs[s1col + mr][kb + 1];

            c0 = __builtin_amdgcn_wmma_f32_16x16x4_f32(
                    false, a, false, b0, (short)0, c0, false, false);
            c1 = __builtin_amdgcn_wmma_f32_16x16x4_f32(
                    false, a, false, b1, (short)0, c1, false, false);
        }
        __syncthreads();
    }

    // C/D layout: VGPR v: lanes 0-15 -> (M=v,   N=lane),
    //                     lanes16-31 -> (M=v+8, N=lane-16)
    const int nIdx = lane & 15;
    const int mOff = (lane >> 4) << 3;
#pragma unroll
    for (int v = 0; v < 8; ++v) {
        {
            const int row = tm + s0row + mOff + v;
            const int col = tn + s0col + nIdx;
            // sigmoid via v_exp_f32 + v_rcp_f32 (no precise-div sequence)
            out[row * BATCH + col] =
                __builtin_amdgcn_rcpf(1.0f + __expf(-c0[v]));
        }
        {
            const int row = tm + s0row + mOff + v;
            const int col = tn + s1col + nIdx;
            out[row * BATCH + col] =
                __builtin_amdgcn_rcpf(1.0f + __expf(-c1[v]));
        }
    }
}

// ---------------------------------------------------------------------------
extern "C" void kernel_launch(void* const* d_in, const int* in_sizes, int n_in,
                              void* d_out, int out_size, void* d_ws, size_t ws_size,
                              hipStream_t stream) {
    const int*   bh  = (const int*)d_in[0];
    const int*   bt  = (const int*)d_in[1];
    const int*   br  = (const int*)d_in[2];
    const float* ent = (const float*)d_in[3];
    const float* rel = (const float*)d_in[4];
    float*       out = (float*)d_out;

    float* hr    = (float*)d_ws;                 // 2048*256 f32 = 2 MB
    float* tails = hr + (size_t)BATCH * EMB;     // 2048*256 f32 = 2 MB

    hr_gather_kernel<<<BATCH, EMB, 0, stream>>>(bh, bt, br, ent, rel, hr, tails);

    dim3 grid(BATCH / TILE, BATCH / TILE);       // 32 x 32 blocks
    score_wmma_kernel<<<grid, 256, 0, stream>>>(hr, tails, out);
}